// AttnDecoderRNN_274877907472
// MI455X (gfx1250) — compile-verified
//
#include <hip/hip_runtime.h>
#include <hip/hip_bf16.h>
#include <math.h>

// ---------------------------------------------------------------------------
// AttnDecoderRNN single step, fp32, CDNA5 (gfx1250, wave32).
//  * GEMMs: V_WMMA_F32_16X16X4_F32
//  * A-panel staged in LDS with GLOBAL_LOAD_ASYNC_TO_LDS_B128 (ASYNCcnt)
//  * W stream: non-temporal loads, register-pipelined 2 deep
//  * energy GEMM removed algebraically (softmax shift-invariance kills attn_b)
// ---------------------------------------------------------------------------

typedef __attribute__((ext_vector_type(2))) float v2f;
typedef __attribute__((ext_vector_type(8))) float v8f;

static __device__ __forceinline__ v8f wmma4(v2f a, v2f b, v8f c) {
  return __builtin_amdgcn_wmma_f32_16x16x4_f32(false, a, false, b, (short)0, c,
                                               false, false);
}

static __device__ __forceinline__ float sigmoid_f(float x) {
  return 1.0f / (1.0f + expf(-x));
}

constexpr int kB = 64;
constexpr int kH = 1024;
constexpr int kV = 50000;
constexpr int kS = 100;

// ---------------------------------------------------------------------------
// C[64,N] = A[64,K] @ W[N,K]^T + bias, optional tanh.
// Wave: full M=64 x N=16 tile. Block: 8 waves = 128 columns.
// A panel staged in LDS (async, double-buffered KC=64 chunks, stride 68 ->
// conflict-free ds_load_b64 fragment reads). W streamed NT from HBM with a
// depth-2 register pipeline. Requires N%16==0, K%64==0.
// ---------------------------------------------------------------------------
__global__ __launch_bounds__(256) void gemm_m64_wmma(
    const float* __restrict__ A, const float* __restrict__ W,
    const float* __restrict__ bias, float* __restrict__ C,
    int N, int K, int act_tanh) {
  constexpr int KC = 64;          // K-chunk staged in LDS
  constexpr int LS = 68;          // LDS row stride (dwords): bank-conflict-free
  __shared__ float As[2][64 * LS];

  const int tid  = threadIdx.x;
  const int wave = tid >> 5;
  const int lane = tid & 31;
  int n0 = (blockIdx.x * 8 + wave) * 16;
  if (n0 > N - 16) n0 = N - 16;   // clamp tail: duplicate work, barriers stay uniform
  const int row = lane & 15;
  const int kk  = (lane >> 4) << 1;           // K slot 0 or 2 per half-wave

  const float* wp = W + (size_t)(n0 + row) * K + kk;  // B frag: B[k][n] = W[n][k]

  // async-copy chunk [kc, kc+KC) of the shared A panel into As[buf]
  auto stage = [&](int buf, int kc) {
#pragma unroll
    for (int i = 0; i < 4; ++i) {
      const int q  = i * 256 + tid;           // 1024 float4 tasks: 64 rows x 16
      const int m  = q >> 4;
      const int x4 = q & 15;
      const float* gp = A + (size_t)m * K + kc + x4 * 4;
      const uint32_t ldso =
          (uint32_t)(size_t)(const void*)&As[buf][m * LS + x4 * 4];
      asm volatile("global_load_async_to_lds_b128 %0, %1, off"
                   :: "v"(ldso), "v"(gp) : "memory");
    }
  };

  stage(0, 0);
  asm volatile("s_wait_asynccnt 0x0" ::: "memory");
  __syncthreads();

  v8f c0 = {}, c1 = {}, c2 = {}, c3 = {};
  // W-fragment register pipeline, depth 2 (absolute k indexing, NT stream)
  v2f w_a = __builtin_nontemporal_load((const v2f*)(wp + 0));
  v2f w_b = __builtin_nontemporal_load((const v2f*)(wp + 4));

  for (int kc = 0; kc < K; kc += KC) {
    const int buf = (kc / KC) & 1;
    if (kc + KC < K) stage(buf ^ 1, kc + KC);
    const float* as = &As[buf][0];
#pragma unroll 4
    for (int k2 = 0; k2 < KC; k2 += 4) {
      int kpre = kc + k2 + 8;
      if (kpre >= K) kpre = 0;                // harmless in-bounds dummy
      const v2f w_c = w_a;
      w_a = w_b;
      w_b = __builtin_nontemporal_load((const v2f*)(wp + kpre));
      const int ac = k2 + kk;
      v2f a0 = *(const v2f*)(as + (row     ) * LS + ac);
      v2f a1 = *(const v2f*)(as + (row + 16) * LS + ac);
      v2f a2 = *(const v2f*)(as + (row + 32) * LS + ac);
      v2f a3 = *(const v2f*)(as + (row + 48) * LS + ac);
      c0 = wmma4(a0, w_c, c0);
      c1 = wmma4(a1, w_c, c1);
      c2 = wmma4(a2, w_c, c2);
      c3 = wmma4(a3, w_c, c3);
    }
    asm volatile("s_wait_asynccnt 0x0" ::: "memory");
    __syncthreads();                          // next buffer ready, old one free
  }

  // C/D layout: VGPR i -> (M = i + 8*(lane>=16), N = lane&15)
  const int col = lane & 15;
  const int rb  = (lane >> 4) << 3;
  const float bv = bias ? bias[n0 + col] : 0.0f;
#pragma unroll
  for (int i = 0; i < 8; ++i) {
    float v0 = c0[i] + bv, v1 = c1[i] + bv, v2 = c2[i] + bv, v3 = c3[i] + bv;
    if (act_tanh) { v0 = tanhf(v0); v1 = tanhf(v1); v2 = tanhf(v2); v3 = tanhf(v3); }
    C[(size_t)( 0 + rb + i) * N + n0 + col] = v0;
    C[(size_t)(16 + rb + i) * N + n0 + col] = v1;
    C[(size_t)(32 + rb + i) * N + n0 + col] = v2;
    C[(size_t)(48 + rb + i) * N + n0 + col] = v3;
  }
}

// ---------------------------------------------------------------------------
// u[64,N] = A[64,K] @ W[K,N]   (NN GEMM, W row-major [K,N]; no bias/act).
// Tiny (attn_W is 4 MB, L2-hot) -> plain global loads, same wave tiling.
// ---------------------------------------------------------------------------
__global__ __launch_bounds__(256) void gemm_m64_nn_wmma(
    const float* __restrict__ A, const float* __restrict__ W,
    float* __restrict__ C, int N, int K) {
  const int wave = threadIdx.x >> 5;
  const int lane = threadIdx.x & 31;
  const int n0 = (blockIdx.x * 8 + wave) * 16;
  if (n0 >= N) return;
  const int row = lane & 15;
  const int kk  = (lane >> 4) << 1;

  const float* ap = A + (size_t)row * K + kk;
  const float* bp = W + (size_t)kk * N + n0 + row;  // B[k][n] = W[k*N + n]

  v8f c0 = {}, c1 = {}, c2 = {}, c3 = {};
  for (int k = 0; k < K; k += 4) {
    v2f b;
    b.x = bp[(size_t)k * N];
    b.y = bp[(size_t)k * N + N];
    v2f a0 = *(const v2f*)(ap + k);
    v2f a1 = *(const v2f*)(ap + (size_t)16 * K + k);
    v2f a2 = *(const v2f*)(ap + (size_t)32 * K + k);
    v2f a3 = *(const v2f*)(ap + (size_t)48 * K + k);
    c0 = wmma4(a0, b, c0);
    c1 = wmma4(a1, b, c1);
    c2 = wmma4(a2, b, c2);
    c3 = wmma4(a3, b, c3);
  }

  const int col = lane & 15;
  const int rb  = (lane >> 4) << 3;
#pragma unroll
  for (int i = 0; i < 8; ++i) {
    C[(size_t)( 0 + rb + i) * N + n0 + col] = c0[i];
    C[(size_t)(16 + rb + i) * N + n0 + col] = c1[i];
    C[(size_t)(32 + rb + i) * N + n0 + col] = c2[i];
    C[(size_t)(48 + rb + i) * N + n0 + col] = c3[i];
  }
}

// ---------------------------------------------------------------------------
// Elementwise / reduction kernels
// ---------------------------------------------------------------------------
__global__ void gather_emb(const int* __restrict__ seq,
                           const float* __restrict__ emb,
                           float* __restrict__ out) {
  const int b = blockIdx.x;
  const float4* src = (const float4*)(emb + (size_t)seq[b] * kH);
  float4* dst = (float4*)(out + (size_t)b * kH);
  dst[threadIdx.x] = src[threadIdx.x];
}

// PyTorch GRU gate order r,z,n. gi/gh are [B,3H]; hprev,hnew are [B,H].
__global__ void gru_pw(const float* __restrict__ gi, const float* __restrict__ gh,
                       const float* __restrict__ hprev, float* __restrict__ hnew) {
  const int idx = blockIdx.x * blockDim.x + threadIdx.x;
  const int b = idx >> 10, h = idx & (kH - 1);
  const float* gib = gi + (size_t)b * 3 * kH;
  const float* ghb = gh + (size_t)b * 3 * kH;
  const float r = sigmoid_f(gib[h] + ghb[h]);
  const float z = sigmoid_f(gib[kH + h] + ghb[kH + h]);
  const float n = tanhf(gib[2 * kH + h] + r * ghb[2 * kH + h]);
  hnew[idx] = (1.0f - z) * n + z * hprev[idx];
}

// scores[b,s] = sum_k u[b,k] * enc[s,b,k]   (u = h1 @ attn_W; attn_b cancels
// in softmax). grid (S, B), 256 threads reduce a 1024-length dot.
__global__ __launch_bounds__(256) void scores_k(
    const float* __restrict__ u, const float* __restrict__ enc,
    float* __restrict__ scores) {
  const int s = blockIdx.x, b = blockIdx.y, t = threadIdx.x;
  const float4 ev = ((const float4*)(enc + ((size_t)s * kB + b) * kH))[t];
  const float4 uv = ((const float4*)(u + (size_t)b * kH))[t];
  float p = ev.x * uv.x + ev.y * uv.y + ev.z * uv.z + ev.w * uv.w;
  __shared__ float red[256];
  red[t] = p;
  __syncthreads();
  for (int off = 128; off > 0; off >>= 1) {
    if (t < off) red[t] += red[t + off];
    __syncthreads();
  }
  if (t == 0) scores[b * kS + s] = red[0];
}

__global__ __launch_bounds__(128) void softmax_k(
    const float* __restrict__ scores, float* __restrict__ attnw) {
  const int b = blockIdx.x, t = threadIdx.x;
  __shared__ float red[128];
  const float v = (t < kS) ? scores[b * kS + t] : -INFINITY;
  red[t] = v;
  __syncthreads();
  for (int off = 64; off > 0; off >>= 1) {
    if (t < off) red[t] = fmaxf(red[t], red[t + off]);
    __syncthreads();
  }
  const float mx = red[0];
  __syncthreads();
  const float e = (t < kS) ? expf(v - mx) : 0.0f;
  red[t] = e;
  __syncthreads();
  for (int off = 64; off > 0; off >>= 1) {
    if (t < off) red[t] += red[t + off];
    __syncthreads();
  }
  if (t < kS) attnw[b * kS + t] = e / red[0];
}

// context[b,h] = sum_s w[b,s] * enc[s,b,h]
__global__ void context_k(const float* __restrict__ attnw,
                          const float* __restrict__ enc,
                          float* __restrict__ ctx) {
  const int idx = blockIdx.x * blockDim.x + threadIdx.x;
  const int b = idx >> 10, h = idx & (kH - 1);
  float acc = 0.0f;
  for (int s = 0; s < kS; ++s)
    acc = fmaf(attnw[b * kS + s], enc[((size_t)s * kB + b) * kH + h], acc);
  ctx[idx] = acc;
}

__global__ void concat_k(const float* __restrict__ h1, const float* __restrict__ ctx,
                         float* __restrict__ cat) {
  const int idx = blockIdx.x * blockDim.x + threadIdx.x;  // B*2H
  const int b = idx >> 11, j = idx & (2 * kH - 1);
  cat[idx] = (j < kH) ? h1[(size_t)b * kH + j] : ctx[(size_t)b * kH + (j - kH)];
}

__global__ void copy_hidden(const float* __restrict__ h0,
                            const float* __restrict__ h1,
                            float* __restrict__ dst) {
  const int idx = blockIdx.x * blockDim.x + threadIdx.x;
  dst[idx] = h0[idx];
  dst[kB * kH + idx] = h1[idx];
}

// ---------------------------------------------------------------------------
// Launcher
// ---------------------------------------------------------------------------
extern "C" void kernel_launch(void* const* d_in, const int* in_sizes, int n_in,
                              void* d_out, int out_size, void* d_ws, size_t ws_size,
                              hipStream_t stream) {
  (void)in_sizes; (void)n_in; (void)out_size; (void)ws_size;

  const int*   seq   = (const int*)  d_in[0];
  const float* lasth = (const float*)d_in[1];   // [2, B, H]
  const float* enc   = (const float*)d_in[2];   // [S, B, H]
  const float* emb   = (const float*)d_in[3];   // [V, H]
  const float* Wih0  = (const float*)d_in[4];
  const float* Whh0  = (const float*)d_in[5];
  const float* bih0  = (const float*)d_in[6];
  const float* bhh0  = (const float*)d_in[7];
  const float* Wih1  = (const float*)d_in[8];
  const float* Whh1  = (const float*)d_in[9];
  const float* bih1  = (const float*)d_in[10];
  const float* bhh1  = (const float*)d_in[11];
  const float* attnW = (const float*)d_in[12];  // [H, H]
  const float* catW  = (const float*)d_in[14];  // [H, 2H]
  const float* catb  = (const float*)d_in[15];
  const float* outW  = (const float*)d_in[16];  // [V, H]
  const float* outb  = (const float*)d_in[17];

  float* out = (float*)d_out;
  float* out_logits = out;                          // [B, V]
  float* out_hidden = out + (size_t)kB * kV;        // [2, B, H]
  float* out_attnw  = out_hidden + 2 * kB * kH;     // [B, 1, S]

  float* ws       = (float*)d_ws;
  float* embedded = ws;                              // 64*1024
  float* gi       = embedded + kB * kH;              // 64*3072
  float* gh       = gi + kB * 3 * kH;                // 64*3072
  float* h0       = gh + kB * 3 * kH;                // 64*1024
  float* h1       = h0 + kB * kH;                    // 64*1024
  float* u        = h1 + kB * kH;                    // 64*1024
  float* scores   = u + kB * kH;                     // 64*100
  float* ctx      = scores + kB * kS;                // 64*1024
  float* cat      = ctx + kB * kH;                   // 64*2048
  float* cout     = cat + kB * 2 * kH;               // 64*1024

  // 1) embedding gather
  gather_emb<<<kB, 256, 0, stream>>>(seq, emb, embedded);

  // 2) GRU layer 0
  gemm_m64_wmma<<<24, 256, 0, stream>>>(embedded, Wih0, bih0, gi, 3 * kH, kH, 0);
  gemm_m64_wmma<<<24, 256, 0, stream>>>(lasth,    Whh0, bhh0, gh, 3 * kH, kH, 0);
  gru_pw<<<kB * kH / 256, 256, 0, stream>>>(gi, gh, lasth, h0);

  // 3) GRU layer 1
  gemm_m64_wmma<<<24, 256, 0, stream>>>(h0,              Wih1, bih1, gi, 3 * kH, kH, 0);
  gemm_m64_wmma<<<24, 256, 0, stream>>>(lasth + kB * kH, Whh1, bhh1, gh, 3 * kH, kH, 0);
  gru_pw<<<kB * kH / 256, 256, 0, stream>>>(gi, gh, lasth + kB * kH, h1);

  // 4) attention: u = h1 @ attn_W ; scores = <u, enc> ; softmax ; context
  gemm_m64_nn_wmma<<<kH / 128, 256, 0, stream>>>(h1, attnW, u, kH, kH);
  {
    dim3 g(kS, kB);
    scores_k<<<g, 256, 0, stream>>>(u, enc, scores);
  }
  softmax_k<<<kB, 128, 0, stream>>>(scores, out_attnw);
  context_k<<<kB * kH / 256, 256, 0, stream>>>(out_attnw, enc, ctx);

  // 5) concat + tanh(Linear)  (K=2048)
  concat_k<<<kB * 2 * kH / 256, 256, 0, stream>>>(h1, ctx, cat);
  gemm_m64_wmma<<<8, 256, 0, stream>>>(cat, catW, catb, cout, kH, 2 * kH, 1);

  // 6) logits = cout @ out_W^T + out_b  (out_W streamed once, NT)
  gemm_m64_wmma<<<(kV + 127) / 128, 256, 0, stream>>>(cout, outW, outb,
                                                      out_logits, kV, kH, 0);

  // 7) hidden = stack(h0, h1)
  copy_hidden<<<kB * kH / 256, 256, 0, stream>>>(h0, h1, out_hidden);
}